// RecurrentGCN_54305566491125
// MI455X (gfx1250) — compile-verified
//
#include <hip/hip_runtime.h>

typedef __attribute__((ext_vector_type(2))) float v2f;
typedef __attribute__((ext_vector_type(8))) float v8f;

#define FH 64

// ---------------- prep kernels ----------------

__global__ void zero_f32_kernel(float* __restrict__ p, long n) {
  long i = (long)blockIdx.x * blockDim.x + threadIdx.x;
  long stride = (long)gridDim.x * blockDim.x;
  for (; i < n; i += stride) p[i] = 0.0f;
}

__global__ void deg_kernel(const int* __restrict__ ei, const float* __restrict__ ew,
                           float* __restrict__ deg, int E) {
  int e = blockIdx.x * blockDim.x + threadIdx.x;
  if (e >= E) return;
  int r = ei[e];
  int c = ei[E + e];
  float w = (r == c) ? 0.0f : ew[e];
  if (w != 0.0f) atomicAdd(&deg[r], w);
}

__global__ void dinv_kernel(const float* __restrict__ deg, float* __restrict__ dinv, int N) {
  int i = blockIdx.x * blockDim.x + threadIdx.x;
  if (i >= N) return;
  float d = deg[i];
  dinv[i] = (d > 0.0f) ? rsqrtf(d) : 0.0f;
}

__global__ void norm_kernel(const int* __restrict__ ei, const float* __restrict__ ew,
                            const float* __restrict__ dinv, float* __restrict__ nrm, int E) {
  int e = blockIdx.x * blockDim.x + threadIdx.x;
  if (e >= E) return;
  int r = ei[e];
  int c = ei[E + e];
  float w = (r == c) ? 0.0f : ew[e];
  nrm[e] = -dinv[r] * w * dinv[c];
}

// one wave per edge: 32 lanes x float2 = 64 features, coalesced gather + atomic scatter
__global__ void scatter_kernel(const int* __restrict__ ei, const float* __restrict__ nrm,
                               const float* __restrict__ x, float* __restrict__ tx1, int E) {
  long gid = (long)blockIdx.x * blockDim.x + threadIdx.x;
  int e = (int)(gid >> 5);
  if (e >= E) return;
  int fp = ((int)gid & 31) * 2;
  float nv = nrm[e];
  if (nv == 0.0f) return;
  int r = ei[e];
  int c = ei[E + e];
  v2f xv = *(const v2f*)(x + (long)r * FH + fp);
  float* dst = tx1 + (long)c * FH + fp;
  atomicAdd(dst + 0, nv * xv.x);
  atomicAdd(dst + 1, nv * xv.y);
}

// ---------------- fused WMMA kernel ----------------
// One wave (32 threads) per 16-node tile.
// A-frag layout (V_WMMA_F32_16X16X4_F32): lane l<16 -> M=l holds K={kb,kb+1};
// lane l>=16 -> M=l-16 holds K={kb+2,kb+3}.
// B-frag: lane l holds W[kb+koff][n], W[kb+koff+1][n] with n = l&15, koff = 2*(l>>4).
// C/D: vgpr r, lanes 0-15 -> (M=r, N=l); lanes 16-31 -> (M=r+8, N=l-16).

__launch_bounds__(32)
__global__ void fused_gru_kernel(const float* __restrict__ x, const float* __restrict__ tx1,
                                 const float* __restrict__ Wxz0, const float* __restrict__ Wxz1,
                                 const float* __restrict__ Wxh0, const float* __restrict__ Wxh1,
                                 const float* __restrict__ Wlin,
                                 const float* __restrict__ bxz, const float* __restrict__ bhz,
                                 const float* __restrict__ bxh, const float* __restrict__ bhh,
                                 const float* __restrict__ blin,
                                 float* __restrict__ out, int N) {
  __shared__ float hs[16 * FH];  // relu(H) tile staged through LDS

  const int lane = threadIdx.x & 31;
  const int m = lane & 15;            // A/B row-or-col index for this lane
  const int koff = (lane >> 4) * 2;   // which K pair this half-wave owns
  const int hi = (lane >> 4) * 8;     // C/D row offset for upper half-wave
  const long tbase = (long)blockIdx.x * 16;

  // --- preload fp32 A-fragments of x and tx1 (16 K-steps each) ---
  long node = tbase + m;
  if (node >= N) node = N - 1;  // clamp: clamped rows are never stored
  const float* xrow = x + node * FH;
  const float* trow = tx1 + node * FH;
  v2f ax[16], at[16];
#pragma unroll
  for (int kk = 0; kk < 16; ++kk) {
    int off = kk * 4 + koff;
    ax[kk] = *(const v2f*)(xrow + off);
    at[kk] = *(const v2f*)(trow + off);
  }

  auto ldB = [&](const float* __restrict__ W, int kb, int nc) -> v2f {
    v2f b;
    b.x = W[(kb + koff) * FH + nc];
    b.y = W[(kb + koff + 1) * FH + nc];
    return b;
  };

  // --- Z and Ht paths: 4 output col-tiles, K=64 in 16 steps of 4 ---
#pragma unroll
  for (int nt = 0; nt < 4; ++nt) {
    const int nc = nt * 16 + m;
    v8f accz = {0.f, 0.f, 0.f, 0.f, 0.f, 0.f, 0.f, 0.f};
    v8f acch = {0.f, 0.f, 0.f, 0.f, 0.f, 0.f, 0.f, 0.f};
#pragma unroll
    for (int kk = 0; kk < 16; ++kk) {
      int kb = kk * 4;
      accz = __builtin_amdgcn_wmma_f32_16x16x4_f32(false, ax[kk], false, ldB(Wxz0, kb, nc),
                                                   (short)0, accz, false, false);
      accz = __builtin_amdgcn_wmma_f32_16x16x4_f32(false, at[kk], false, ldB(Wxz1, kb, nc),
                                                   (short)0, accz, false, false);
      acch = __builtin_amdgcn_wmma_f32_16x16x4_f32(false, ax[kk], false, ldB(Wxh0, kb, nc),
                                                   (short)0, acch, false, false);
      acch = __builtin_amdgcn_wmma_f32_16x16x4_f32(false, at[kk], false, ldB(Wxh1, kb, nc),
                                                   (short)0, acch, false, false);
    }
    const float bzv = bxz[nc] + bhz[nc];
    const float bhv = bxh[nc] + bhh[nc];
#pragma unroll
    for (int r = 0; r < 8; ++r) {
      float z = 1.0f / (1.0f + expf(-(accz[r] + bzv)));
      float ht = tanhf(acch[r] + bhv);
      float h = (1.0f - z) * ht;  // H = Z*0 + (1-Z)*Ht
      hs[(r + hi) * FH + nc] = (h > 0.0f) ? h : 0.0f;  // relu(H)
    }
  }

  __syncthreads();  // single-wave block: lowers to LDS-count wait

  // --- final GEMM: relu(H) @ W_lin + b_lin ---
  v2f ah[16];
#pragma unroll
  for (int kk = 0; kk < 16; ++kk) {
    ah[kk] = *(const v2f*)(&hs[m * FH + kk * 4 + koff]);
  }
#pragma unroll
  for (int nt = 0; nt < 4; ++nt) {
    const int nc = nt * 16 + m;
    v8f acc = {0.f, 0.f, 0.f, 0.f, 0.f, 0.f, 0.f, 0.f};
#pragma unroll
    for (int kk = 0; kk < 16; ++kk) {
      acc = __builtin_amdgcn_wmma_f32_16x16x4_f32(false, ah[kk], false, ldB(Wlin, kk * 4, nc),
                                                  (short)0, acc, false, false);
    }
    const float bl = blin[nc];
#pragma unroll
    for (int r = 0; r < 8; ++r) {
      long orow = tbase + r + hi;
      if (orow < N) out[orow * FH + nc] = acc[r] + bl;
    }
  }
}

// ---------------- launch ----------------

extern "C" void kernel_launch(void* const* d_in, const int* in_sizes, int n_in,
                              void* d_out, int out_size, void* d_ws, size_t ws_size,
                              hipStream_t stream) {
  const float* x    = (const float*)d_in[0];
  const int*   ei   = (const int*)d_in[1];
  const float* ew   = (const float*)d_in[2];
  const float* Wxz0 = (const float*)d_in[3];
  const float* Wxz1 = (const float*)d_in[4];
  const float* bxz  = (const float*)d_in[5];
  const float* bhz  = (const float*)d_in[8];
  const float* Wxh0 = (const float*)d_in[15];
  const float* Wxh1 = (const float*)d_in[16];
  const float* bxh  = (const float*)d_in[17];
  const float* bhh  = (const float*)d_in[20];
  const float* Wlin = (const float*)d_in[21];
  const float* blin = (const float*)d_in[22];

  const int N = in_sizes[0] / FH;
  const int E = in_sizes[1] / 2;

  // workspace layout (floats): [deg: N][tx1: N*64][dinv: N][norm: E]
  float* ws   = (float*)d_ws;
  float* deg  = ws;
  float* tx1  = ws + N;
  float* dinv = tx1 + (size_t)N * FH;
  float* nrm  = dinv + N;

  zero_f32_kernel<<<1024, 256, 0, stream>>>(deg, (long)N * (FH + 1));  // deg + tx1 contiguous
  deg_kernel<<<(E + 255) / 256, 256, 0, stream>>>(ei, ew, deg, E);
  dinv_kernel<<<(N + 255) / 256, 256, 0, stream>>>(deg, dinv, N);
  norm_kernel<<<(E + 255) / 256, 256, 0, stream>>>(ei, ew, dinv, nrm, E);

  long sthreads = (long)E * 32;
  scatter_kernel<<<(int)((sthreads + 255) / 256), 256, 0, stream>>>(ei, nrm, x, tx1, E);

  fused_gru_kernel<<<(N + 15) / 16, 32, 0, stream>>>(x, tx1, Wxz0, Wxz1, Wxh0, Wxh1, Wlin,
                                                     bxz, bhz, bxh, bhh, blin,
                                                     (float*)d_out, N);
}